// LocallyConnected1D_3710851744183
// MI455X (gfx1250) — compile-verified
//
#include <hip/hip_runtime.h>

typedef __attribute__((ext_vector_type(2))) float v2f;
typedef __attribute__((ext_vector_type(4))) float v4f;
typedef __attribute__((ext_vector_type(8))) float v8f;
typedef __attribute__((ext_vector_type(4))) unsigned int v4u;
typedef __attribute__((ext_vector_type(8))) unsigned int v8u;

#define B_  32
#define S_  4096
#define C_  64
#define F_  64
#define KC_ 192      // KSIZE*C = 3*64
#define L_  4094

#define A_STRIDE 196 // 192 + 4 pad: conflict-free b64 A-fragment reads
#define B_STRIDE 72  // 64 + 8 pad: produced by TDM pad_interval=5 / pad_amount=7

__global__ __launch_bounds__(256)
void lc1d_wmma_f32(const float* __restrict__ x, const float* __restrict__ w,
                   const float* __restrict__ bias, float* __restrict__ out)
{
    __shared__ __attribute__((aligned(16))) float Asm[B_ * A_STRIDE];   // 25,088 B
    __shared__ __attribute__((aligned(16))) float Bsm[KC_ * B_STRIDE];  // 55,296 B

    const int l   = blockIdx.x;
    const int tid = threadIdx.x;

    // ---- Stage W[l] (192x64 f32 = 48 KB) via Tensor Data Mover: one DMA,
    //      no VGPR round-trip; HW pads 8 DWORDs per 64-DWORD row -> stride 72.
    if (tid < 32) {   // wave 0 issues; EXEC is ignored by TDM, descriptor in SGPRs
        const unsigned long long ga =
            (unsigned long long)(const void*)(w + (size_t)l * (KC_ * F_));
        const unsigned int lds_b = (unsigned int)(size_t)(void*)&Bsm[0];

        // D# group 0: count=1 | lds_addr | global_addr[56:0] | type=2
        v4u g0;
        g0.x = 1u;                                   // count=1, is_restore=0
        g0.y = lds_b;                                // LDS byte address
        g0.z = (unsigned int)(ga & 0xffffffffull);   // global_addr[31:0]
        g0.w = (unsigned int)(ga >> 32) | (2u << 30);// global_addr[56:32] | type=2

        // D# group 1
        v8u g1;
        g1.s0 = (2u << 16)      // data_size = 4 bytes
              | (1u << 20)      // pad_enable
              | (5u << 22)      // pad_interval: 2^(5+1) = 64 DWORDs (one F-row)
              | (7u << 25);     // pad_amount:  7+1 = 8 DWORDs
        g1.s1 = (64u & 0xffffu) << 16;               // tensor_dim0 = 64 (low16)
        g1.s2 = (64u >> 16) | ((192u & 0xffffu) << 16); // dim0 hi | tensor_dim1=192 lo
        g1.s3 = (192u >> 16) | (64u << 16);          // dim1 hi | tile_dim0 = 64
        g1.s4 = 192u;                                // tile_dim1 = 192, tile_dim2 = 0
        g1.s5 = 64u;                                 // tensor_dim0_stride = 64 (lo32)
        g1.s6 = (0u) | ((12288u & 0xffffu) << 16);   // stride0 hi | tensor_dim1_stride lo16
        g1.s7 = 12288u >> 16;                        // tensor_dim1_stride hi

        asm volatile("tensor_load_to_lds %0, %1"
                     :
                     : "s"(g0), "s"(g1)
                     : "memory");
    }

    // ---- Stage patches x[b, l:l+3, :] (32x192 f32, contiguous per row; L2-resident) ----
    {
        #pragma unroll
        for (int it = 0; it < (B_ * KC_) / (4 * 256); ++it) {   // 6 iters
            const int i   = it * 256 + tid;
            const int row = (i * 4) / KC_;     // b
            const int col = (i * 4) % KC_;     // kc (multiple of 4)
            const v4f v = *(const v4f*)(x + ((size_t)row * S_ + l) * C_ + col);
            float* dst = &Asm[row * A_STRIDE + col];
            dst[0] = v.x; dst[1] = v.y; dst[2] = v.z; dst[3] = v.w;
        }
    }

    if (tid < 32) {
        __builtin_amdgcn_s_wait_tensorcnt(0);   // TDM tile landed in LDS
    }
    __syncthreads();

    // ---- 8 waves: wave -> (m-tile, n-tile) of the 32x64 output tile ----
    const int wave = tid >> 5;         // wave32
    const int lane = tid & 31;
    const int mt   = wave >> 2;        // 0..1  (batch rows mt*16..+15)
    const int nt   = wave & 3;         // 0..3  (filter cols nt*16..+15)
    const int hi   = lane >> 4;        // half-wave selects K pair per ISA layout
    const int l16  = lane & 15;

    const float* Arow = &Asm[(mt * 16 + l16) * A_STRIDE];
    const int    nloc = nt * 16 + l16;

    v8f acc = {};
    #pragma unroll 4
    for (int k0 = 0; k0 < KC_; k0 += 4) {
        const int ka = k0 + hi * 2;
        // A 16x4 fragment: lane row = m, VGPR0/1 = K=(ka, ka+1)   (8B-aligned b64)
        v2f a = *(const v2f*)(Arow + ka);
        // B 4x16 fragment: lane col = n, VGPR0/1 = K=(ka, ka+1)
        v2f b;
        b.x = Bsm[ka * B_STRIDE + nloc];
        b.y = Bsm[(ka + 1) * B_STRIDE + nloc];
        acc = __builtin_amdgcn_wmma_f32_16x16x4_f32(
                  /*neg_a=*/false, a, /*neg_b=*/false, b,
                  /*c_mod=*/(short)0, acc, /*reuse_a=*/false, /*reuse_b=*/false);
    }

    // ---- Epilogue: bias + relu, C/D layout: VGPR i -> M = i + 8*hi, N = l16 ----
    const float bv    = bias[(size_t)l * F_ + nloc];
    float*      obase = out + (size_t)l * F_ + nloc;   // out[b][l][f]
    const int   mbase = mt * 16 + hi * 8;
    #pragma unroll
    for (int i = 0; i < 8; ++i) {
        float v = acc[i] + bv;
        v = v > 0.f ? v : 0.f;
        obase[(size_t)(mbase + i) * (L_ * F_)] = v;
    }
}

extern "C" void kernel_launch(void* const* d_in, const int* in_sizes, int n_in,
                              void* d_out, int out_size, void* d_ws, size_t ws_size,
                              hipStream_t stream) {
    const float* x    = (const float*)d_in[0];
    const float* w    = (const float*)d_in[1];
    const float* bias = (const float*)d_in[2];
    float*       out  = (float*)d_out;

    dim3 grid(L_);
    dim3 block(256);
    hipLaunchKernelGGL(lc1d_wmma_f32, grid, block, 0, stream, x, w, bias, out);
}